// StaticAggregatedFlow_81784767251141
// MI455X (gfx1250) — compile-verified
//
#include <hip/hip_runtime.h>

typedef float v2f __attribute__((ext_vector_type(2)));
typedef float v4f __attribute__((ext_vector_type(4)));
typedef float v8f __attribute__((ext_vector_type(8)));

#define BB 8
#define NP 200000
#define HH 640
#define WW 640
#define HW (HH * WW)
#define OUT_FLOW (BB * 2 * HW)
#define GROUPS_PER_BATCH (NP / 16)   // 12500
#define WS_PER_BATCH 33              // 16 weighted Gram + 16 unweighted Gram + count

// ---------------------------------------------------------------------------
// Kernel 0: zero the workspace accumulators (re-run every graph replay)
// ---------------------------------------------------------------------------
__global__ void saf_init_ws(float* __restrict__ ws) {
  int i = blockIdx.x * blockDim.x + threadIdx.x;
  if (i < BB * WS_PER_BATCH) ws[i] = 0.0f;
}

// ---------------------------------------------------------------------------
// Kernel 1: per-batch 4x4 Gram accumulation via V_WMMA_F32_16X16X4_F32.
// Each WMMA consumes 16 points: A[4g+i][k] = a-component i of point 4g+k.
// Diagonal 4x4 blocks of the 16x16 accumulator hold sum(a_n * b_n^T).
// Branch-free loads; block-level LDS reduction before global atomics.
// ---------------------------------------------------------------------------
__global__ void saf_gram_reduce(const float* __restrict__ static_flow,
                                const float* __restrict__ staticness,
                                const float* __restrict__ pc,
                                const int* __restrict__ coords,
                                const unsigned char* __restrict__ valid,
                                float* __restrict__ ws) {
  __shared__ float lws[WS_PER_BATCH];
  if (threadIdx.x < WS_PER_BATCH) lws[threadIdx.x] = 0.0f;
  __syncthreads();

  const int batch = blockIdx.y;
  const int lane = threadIdx.x & 31;
  const int wavesPerBatch = gridDim.x * (blockDim.x >> 5);
  const int waveId = blockIdx.x * (blockDim.x >> 5) + (threadIdx.x >> 5);

  const int m = lane & 15;         // matrix row (A) / col (B) index
  const int g = m >> 2;            // which 4-point diagonal subgroup
  const int c = m & 3;             // vector component (0..2 = xyz, 3 = homog 1)
  const bool hi = lane >= 16;
  const int k0 = hi ? 2 : 0;       // K indices {k0, k0+1}
  const bool isOne = (c == 3);     // homogeneous-1 lane
  const int cc = isOne ? 2 : c;    // safe pc component to load
  const int fc = c & 1;            // safe flow plane to load

  const float* sflow_b = static_flow + (size_t)batch * 2 * HW;
  const float* stat_b  = staticness + (size_t)batch * HW;
  const float* pc_b    = pc + (size_t)batch * NP * 6;
  const int*   co_b    = coords + (size_t)batch * NP * 2;
  const unsigned char* va_b = valid + (size_t)batch * NP;

  v8f Cw = {};   // weighted Gram accumulator
  v8f Cu = {};   // unweighted Gram accumulator
  int cnt = 0;   // count of (w > 0), counted once per point by c==3 lanes

  for (int grp = waveId; grp < GROUPS_PER_BATCH; grp += wavesPerBatch) {
    const int pbase = grp * 16 + 4 * g + k0;
    float aa[2], bw[2], bu[2];
#pragma unroll
    for (int kk = 0; kk < 2; ++kk) {
      const int p = pbase + kk;
      const int xi = __builtin_nontemporal_load(&co_b[p * 2 + 0]);
      const int yi = __builtin_nontemporal_load(&co_b[p * 2 + 1]);
      const int pix = xi * WW + yi;
      const unsigned char vm = __builtin_nontemporal_load(&va_b[p]);
      const float sv = stat_b[pix];
      const float w = vm ? sv : 0.0f;
      const float p0 = __builtin_nontemporal_load(&pc_b[p * 6 + cc]);
      const float p1 = __builtin_nontemporal_load(&pc_b[p * 6 + 3 + cc]);
      const float flv = sflow_b[fc * HW + pix];
      const float A = p0 + p1;
      const float fl = (c < 2) ? flv : 0.0f;
      const float a = isOne ? 1.0f : A;
      const float b = isOne ? 1.0f : (A + fl);
      cnt += (isOne && w > 0.0f) ? 1 : 0;
      aa[kk] = a;
      bu[kk] = b;
      bw[kk] = w * b;
    }
    v2f av  = {aa[0], aa[1]};
    v2f bwv = {bw[0], bw[1]};
    v2f buv = {bu[0], bu[1]};
    Cw = __builtin_amdgcn_wmma_f32_16x16x4_f32(false, av, false, bwv,
                                               (short)0, Cw, false, false);
    Cu = __builtin_amdgcn_wmma_f32_16x16x4_f32(false, av, false, buv,
                                               (short)0, Cu, false, false);
  }

  // Harvest the diagonal 4x4 blocks into LDS: lane holds D[r + (hi?8:0)][n].
  const int n = lane & 15;
  const int gn = n >> 2;
  const int j = n & 3;
#pragma unroll
  for (int r = 0; r < 8; ++r) {
    const int mr = r + (hi ? 8 : 0);
    if ((mr >> 2) == gn) {
      const int i = mr & 3;
      atomicAdd(&lws[i * 4 + j], Cw[r]);
      atomicAdd(&lws[16 + i * 4 + j], Cu[r]);
    }
  }
  if (isOne && cnt > 0) atomicAdd(&lws[32], (float)cnt);
  __syncthreads();

  // One global atomic per Gram entry per block.
  if (threadIdx.x < WS_PER_BATCH)
    atomicAdd(&ws[batch * WS_PER_BATCH + threadIdx.x], lws[threadIdx.x]);
}

// ---------------------------------------------------------------------------
// Kernel 2: per-batch finalize. Gram -> Hc -> R = polar(Hc^T) via Newton,
// assemble 4x4 T, write to the T section of d_out.
// ---------------------------------------------------------------------------
__global__ void saf_finalize(const float* __restrict__ ws, float* __restrict__ outT) {
  const int b = threadIdx.x;
  if (b >= BB) return;
  const float* wsb = ws + b * WS_PER_BATCH;

  const float cnt = wsb[32];
  const float e = (cnt < 2.5f) ? 1.1920929e-7f : 0.0f;  // FLT_EPSILON correction
  float G[16];
#pragma unroll
  for (int k = 0; k < 16; ++k) G[k] = wsb[k] + e * wsb[16 + k];

  const float Sw = G[15];
  float am[3], bm[3];
#pragma unroll
  for (int i = 0; i < 3; ++i) am[i] = G[i * 4 + 3] / Sw;
#pragma unroll
  for (int j = 0; j < 3; ++j) bm[j] = G[12 + j] / Sw;

  // X = Hc^T where Hc[i][j] = G[i][j]/Sw - am[i]*bm[j]
  float X[9];
#pragma unroll
  for (int i = 0; i < 3; ++i)
#pragma unroll
    for (int j = 0; j < 3; ++j)
      X[j * 3 + i] = G[i * 4 + j] / Sw - am[i] * bm[j];

  // Newton iteration for the orthogonal polar factor: X <- 0.5*(X + X^{-T})
  // X^{-T} = cof(X) / det(X). Converges to R = V*U^T of Hc's SVD.
  for (int it = 0; it < 40; ++it) {
    float C[9];
    C[0] =  X[4] * X[8] - X[5] * X[7];
    C[1] = -(X[3] * X[8] - X[5] * X[6]);
    C[2] =  X[3] * X[7] - X[4] * X[6];
    C[3] = -(X[1] * X[8] - X[2] * X[7]);
    C[4] =  X[0] * X[8] - X[2] * X[6];
    C[5] = -(X[0] * X[7] - X[1] * X[6]);
    C[6] =  X[1] * X[5] - X[2] * X[4];
    C[7] = -(X[0] * X[5] - X[2] * X[3]);
    C[8] =  X[0] * X[4] - X[1] * X[3];
    const float det = X[0] * C[0] + X[1] * C[1] + X[2] * C[2];
    const float s = 0.5f / det;
#pragma unroll
    for (int q = 0; q < 9; ++q) X[q] = 0.5f * X[q] + s * C[q];
  }

  float t[3];
#pragma unroll
  for (int i = 0; i < 3; ++i)
    t[i] = bm[i] - (X[i * 3 + 0] * am[0] + X[i * 3 + 1] * am[1] + X[i * 3 + 2] * am[2]);

  float* Tb = outT + b * 16;
#pragma unroll
  for (int i = 0; i < 3; ++i) {
#pragma unroll
    for (int j = 0; j < 3; ++j) Tb[i * 4 + j] = X[i * 3 + j];
    Tb[i * 4 + 3] = t[i];
  }
  Tb[12] = 0.0f; Tb[13] = 0.0f; Tb[14] = 0.0f; Tb[15] = 1.0f;
}

// ---------------------------------------------------------------------------
// Kernel 3: flow field einsum via WMMA.
// D[16x16] = grid_tile[16x4] x Mt[4x16], columns = 8 batches x 2 components.
// One WMMA produces 16 pixels x all 16 output channels; NT streaming stores.
// ---------------------------------------------------------------------------
__global__ void saf_flow_field(const float* __restrict__ vc,
                               const float* __restrict__ outT,
                               float* __restrict__ out) {
  const int lane = threadIdx.x & 31;
  const bool hi = lane >= 16;
  const int wavesTotal = gridDim.x * (blockDim.x >> 5);
  const int waveId = blockIdx.x * (blockDim.x >> 5) + (threadIdx.x >> 5);

  const int n = lane & 15;       // output channel column
  const int b = n >> 1;
  const int ci = n & 1;
  const int k0 = hi ? 2 : 0;

  // B fragment: column n holds M[b][ci][k] = (T - I)[b][ci][k], loop-invariant.
  const float m0 = outT[b * 16 + ci * 4 + k0]     - ((ci == k0)     ? 1.0f : 0.0f);
  const float m1 = outT[b * 16 + ci * 4 + k0 + 1] - ((ci == k0 + 1) ? 1.0f : 0.0f);
  const v2f bfrag = {m0, m1};
  float* plane = out + ((size_t)b * 2 + ci) * HW;

  const int NGRP = HW / 16;      // 25600 pixel groups
  for (int grp = waveId; grp < NGRP; grp += wavesTotal) {
    // Branch-free A fragment: lo lanes use grid x,y; hi lanes the constants 0,1.
    const int p = grp * 16 + n;
    const float gx = vc[p * 2 + 0];
    const float gy = vc[p * 2 + 1];
    v2f afrag;
    afrag.x = hi ? 0.0f : gx;
    afrag.y = hi ? 1.0f : gy;

    v8f D = {};
    D = __builtin_amdgcn_wmma_f32_16x16x4_f32(false, afrag, false, bfrag,
                                              (short)0, D, false, false);
    const int pbase = grp * 16 + (hi ? 8 : 0);
    v4f lo4 = {D[0], D[1], D[2], D[3]};
    v4f hi4 = {D[4], D[5], D[6], D[7]};
    __builtin_nontemporal_store(lo4, (v4f*)(plane + pbase));
    __builtin_nontemporal_store(hi4, (v4f*)(plane + pbase + 4));
  }
}

// ---------------------------------------------------------------------------
extern "C" void kernel_launch(void* const* d_in, const int* in_sizes, int n_in,
                              void* d_out, int out_size, void* d_ws, size_t ws_size,
                              hipStream_t stream) {
  (void)in_sizes; (void)n_in; (void)out_size; (void)ws_size;
  const float* static_flow = (const float*)d_in[0];          // (B,2,H,W) f32
  const float* staticness  = (const float*)d_in[1];          // (B,1,H,W) f32
  const float* pc          = (const float*)d_in[2];          // (B,N,6)   f32
  const int*   coords      = (const int*)d_in[3];            // (B,N,2)   i32
  const unsigned char* valid = (const unsigned char*)d_in[4];// (B,N)     bool
  const float* vc          = (const float*)d_in[5];          // (H,W,2)   f32

  float* out  = (float*)d_out;
  float* outT = out + OUT_FLOW;   // T (B,4,4) lives after the flow field
  float* ws   = (float*)d_ws;

  saf_init_ws<<<1, 320, 0, stream>>>(ws);
  dim3 grid1(112, BB);
  saf_gram_reduce<<<grid1, 256, 0, stream>>>(static_flow, staticness, pc,
                                             coords, valid, ws);
  saf_finalize<<<1, 32, 0, stream>>>(ws, outT);
  saf_flow_field<<<200, 256, 0, stream>>>(vc, outT, out);
}